// GMM_23278722744337
// MI455X (gfx1250) — compile-verified
//
#include <hip/hip_runtime.h>
#include <hip/hip_bf16.h>
#include <math.h>

typedef __attribute__((ext_vector_type(2))) float v2f;
typedef __attribute__((ext_vector_type(4))) float v4f;
typedef __attribute__((ext_vector_type(8))) float v8f;

#define GN 32768
#define GK 16
#define GD 128
#define GJITTER 1e-6f
#define GLOG2PI 1.8378770664093453f

// Workspace layout (float offsets)
#define WS_GS     0          // 16   : gamma_sum[k]
#define WS_C      32         // 16   : c_k = log(phi) - 0.5*(D*log2pi + logdet)
#define WS_MU     64         // 2048 : mu[k][d]
#define WS_B      4096       // 2048 : b[k][d] = L^{-1} mu
#define WS_PSTAT  8192       // 64*2064 : per-chunk (gs[16], S1[2048])
#define WS_S2P    262144     // 8 splits * K*D*D : partial S2
#define WS_MINVT  2359296    // K*D*D : (L^{-1})^T per k, row e, col d
#define WS_LOGITS 2621440    // N*K
// total end = 3145728 floats = 12.6 MB

// ---------------- Kernel 1: partial stats (gamma_sum, S1 = gamma^T Z) ----------
__global__ __launch_bounds__(256) void k_stats_partial(const float* __restrict__ z,
                                                       const float* __restrict__ gamma,
                                                       float* __restrict__ ws) {
  __shared__ float sred[2 * 2048];
  __shared__ float gsh[32];
  const int tid = threadIdx.x;
  const int d = tid & 127, rg = tid >> 7;
  const int base = blockIdx.x * 512;

  float acc[16];
#pragma unroll
  for (int j = 0; j < 16; j++) acc[j] = 0.f;
  float gacc = 0.f;

  for (int r = base + rg; r < base + 512; r += 2) {
    float zv = z[r * GD + d];
    const v4f* gp = reinterpret_cast<const v4f*>(gamma + r * GK);
    v4f g0 = gp[0], g1 = gp[1], g2 = gp[2], g3 = gp[3];
#pragma unroll
    for (int j = 0; j < 4; j++) {
      acc[j]      += g0[j] * zv;
      acc[4 + j]  += g1[j] * zv;
      acc[8 + j]  += g2[j] * zv;
      acc[12 + j] += g3[j] * zv;
    }
    if (d < 16) gacc += gamma[r * GK + d];
  }
#pragma unroll
  for (int j = 0; j < 16; j++) sred[rg * 2048 + j * 128 + d] = acc[j];
  if (d < 16) gsh[rg * 16 + d] = gacc;
  __syncthreads();

  float* out = ws + WS_PSTAT + blockIdx.x * 2064;
  if (tid < 16) out[tid] = gsh[tid] + gsh[16 + tid];
  for (int idx = tid; idx < 2048; idx += 256)
    out[16 + idx] = sred[idx] + sred[2048 + idx];
}

// ---------------- Kernel 2: reduce stats -> gamma_sum, mu ----------------------
__global__ __launch_bounds__(256) void k_stats_reduce(float* __restrict__ ws) {
  __shared__ float gsh[16];
  const int tid = threadIdx.x;
  if (tid < 16) {
    float g = 0.f;
    for (int c = 0; c < 64; c++) g += ws[WS_PSTAT + c * 2064 + tid];
    ws[WS_GS + tid] = g;
    gsh[tid] = g;
  }
  __syncthreads();
  for (int idx = tid; idx < 2048; idx += 256) {
    float s = 0.f;
    for (int c = 0; c < 64; c++) s += ws[WS_PSTAT + c * 2064 + 16 + idx];
    ws[WS_MU + idx] = s / gsh[idx >> 7];
  }
}

// ---------------- Kernel 3: S2[k] = Z^T diag(gamma_k) Z via f32 WMMA -----------
// 2048 waves: (kgroup of 4) x (8x8 tiles of DxD) x (8 N-splits)
__global__ __launch_bounds__(256) void k_s2_wmma(const float* __restrict__ z,
                                                 const float* __restrict__ gamma,
                                                 float* __restrict__ ws) {
  const int wid = threadIdx.x >> 5;
  const int gw = blockIdx.x * 8 + wid;
  const int split = gw & 7;
  const int tile  = (gw >> 3) & 63;
  const int kg    = gw >> 9;           // 0..3
  const int mi = tile >> 3, ni = tile & 7;

  const int l = threadIdx.x & 31;
  const int mloc = l & 15;
  const int kb = (l >> 4) << 1;        // A/B K-pair base per half-wave
  const int colA = mi * 16 + mloc;     // A-matrix row (M index)
  const int colB = ni * 16 + mloc;     // B-matrix col (N index)

  v8f acc0 = {}, acc1 = {}, acc2 = {}, acc3 = {};

  const int r0 = split * 4096;
  for (int i0 = r0; i0 < r0 + 4096; i0 += 4) {
    const int ra = i0 + kb, rb = ra + 1;
    v2f za, zb, a;
    za.x = z[ra * GD + colA]; za.y = z[rb * GD + colA];
    zb.x = z[ra * GD + colB]; zb.y = z[rb * GD + colB];
    v4f ga = *reinterpret_cast<const v4f*>(gamma + ra * GK + (kg << 2));
    v4f gb = *reinterpret_cast<const v4f*>(gamma + rb * GK + (kg << 2));

    a.x = ga[0] * za.x; a.y = gb[0] * za.y;
    acc0 = __builtin_amdgcn_wmma_f32_16x16x4_f32(false, a, false, zb, (short)0, acc0, false, false);
    a.x = ga[1] * za.x; a.y = gb[1] * za.y;
    acc1 = __builtin_amdgcn_wmma_f32_16x16x4_f32(false, a, false, zb, (short)0, acc1, false, false);
    a.x = ga[2] * za.x; a.y = gb[2] * za.y;
    acc2 = __builtin_amdgcn_wmma_f32_16x16x4_f32(false, a, false, zb, (short)0, acc2, false, false);
    a.x = ga[3] * za.x; a.y = gb[3] * za.y;
    acc3 = __builtin_amdgcn_wmma_f32_16x16x4_f32(false, a, false, zb, (short)0, acc3, false, false);
  }

  // C/D layout: VGPR v -> row (v + 8*(lane>=16)), col = lane&15
  float* dst = ws + WS_S2P + (size_t)split * (GK * GD * GD) + (size_t)(kg * 4) * (GD * GD);
  const int rowBase = mi * 16 + ((l >> 4) << 3);
#pragma unroll
  for (int v = 0; v < 8; v++) {
    const int off = (rowBase + v) * GD + colB;
    dst[off]                  = acc0[v];
    dst[1 * GD * GD + off]    = acc1[v];
    dst[2 * GD * GD + off]    = acc2[v];
    dst[3 * GD * GD + off]    = acc3[v];
  }
}

// ---------------- Kernel 4: sigma, Cholesky, b = L^{-1}mu, MinvT, c_k ----------
__global__ __launch_bounds__(128) void k_chol(float* __restrict__ ws) {
  __shared__ float sig[GD * GD];   // 64 KB
  const int k = blockIdx.x, tid = threadIdx.x;
  const float gs = ws[WS_GS + k];
  const float* mu = ws + WS_MU + k * GD;
  const float* s2 = ws + WS_S2P + (size_t)k * (GD * GD);
  const float mut = mu[tid];

  for (int e = 0; e < GD; e++) {
    float s = 0.f;
#pragma unroll
    for (int sp = 0; sp < 8; sp++)
      s += s2[(size_t)sp * (GK * GD * GD) + tid * GD + e];
    s = s / gs - mut * mu[e];
    if (e == tid) s += GJITTER;
    sig[tid * GD + e] = s;
  }
  __syncthreads();

  // right-looking Cholesky (lower triangle in place)
  for (int j = 0; j < GD; j++) {
    if (tid == 0) sig[j * GD + j] = sqrtf(sig[j * GD + j]);
    __syncthreads();
    const float ljj = sig[j * GD + j];
    if (tid > j) sig[tid * GD + j] /= ljj;
    __syncthreads();
    if (tid > j) {
      const float lij = sig[tid * GD + j];
      for (int e = j + 1; e <= tid; e++)
        sig[tid * GD + e] -= lij * sig[e * GD + j];
    }
    __syncthreads();
  }

  if (tid == 0) {
    float logdet = 0.f;
    for (int j = 0; j < GD; j++) logdet += logf(sig[j * GD + j]);
    logdet *= 2.f;
    ws[WS_C + k] = logf(gs / (float)GN) - 0.5f * ((float)GD * GLOG2PI + logdet);
    // forward solve L y = mu  ->  b_k
    float y[GD];
    for (int j = 0; j < GD; j++) {
      float s = mu[j];
      for (int t = 0; t < j; t++) s -= sig[j * GD + t] * y[t];
      y[j] = s / sig[j * GD + j];
      ws[WS_B + k * GD + j] = y[j];
    }
  }

  // triangular inverse: thread j owns column j of L^{-1}; store transposed
  {
    const int j = tid;
    float col[GD];
    col[j] = 1.f / sig[j * GD + j];
    for (int i = j + 1; i < GD; i++) {
      float s = 0.f;
      for (int t = j; t < i; t++) s += sig[i * GD + t] * col[t];
      col[i] = -s / sig[i * GD + i];
    }
    float* mt = ws + WS_MINVT + (size_t)k * (GD * GD) + j * GD;  // MinvT[j][i] = Minv[i][j]
    for (int i = 0; i < j; i++) mt[i] = 0.f;
    for (int i = j; i < GD; i++) mt[i] = col[i];
  }
}

// ---------------- Kernel 5: maha via WMMA: Y = Z * Minv^T, logits --------------
// 32768 waves: (k) x (group of 16 samples)
__global__ __launch_bounds__(256) void k_maha_wmma(const float* __restrict__ z,
                                                   float* __restrict__ ws) {
  const int wid = threadIdx.x >> 5;
  const int gw = blockIdx.x * 8 + wid;
  const int sg = gw & 2047;
  const int k  = gw >> 11;
  const int i0 = sg << 4;

  const int l = threadIdx.x & 31;
  const int mloc = l & 15;
  const int kb = (l >> 4) << 1;

  const float* mt = ws + WS_MINVT + (size_t)k * (GD * GD);
  const float* zrow = z + (size_t)(i0 + mloc) * GD;   // A-matrix row M=mloc

  v8f acc[8] = {};
  for (int e0 = 0; e0 < GD; e0 += 4) {
    v2f a = *reinterpret_cast<const v2f*>(zrow + e0 + kb);
#pragma unroll
    for (int ni = 0; ni < 8; ni++) {
      v2f b;
      b.x = mt[(e0 + kb) * GD + ni * 16 + mloc];
      b.y = mt[(e0 + kb + 1) * GD + ni * 16 + mloc];
      acc[ni] = __builtin_amdgcn_wmma_f32_16x16x4_f32(false, a, false, b, (short)0, acc[ni], false, false);
    }
  }

  float ss[8];
#pragma unroll
  for (int v = 0; v < 8; v++) ss[v] = 0.f;
#pragma unroll
  for (int ni = 0; ni < 8; ni++) {
    const float bv = ws[WS_B + k * GD + ni * 16 + mloc];
#pragma unroll
    for (int v = 0; v < 8; v++) {
      const float t = acc[ni][v] - bv;
      ss[v] += t * t;
    }
  }
  // reduce across the 16 lanes of each half-wave (n dimension)
#pragma unroll
  for (int v = 0; v < 8; v++) {
    ss[v] += __shfl_xor(ss[v], 1, 32);
    ss[v] += __shfl_xor(ss[v], 2, 32);
    ss[v] += __shfl_xor(ss[v], 4, 32);
    ss[v] += __shfl_xor(ss[v], 8, 32);
  }
  if (mloc == 0) {
    const float ck = ws[WS_C + k];
    const int mb = (l >> 4) << 3;
#pragma unroll
    for (int v = 0; v < 8; v++)
      ws[WS_LOGITS + (size_t)(i0 + mb + v) * GK + k] = ck - 0.5f * ss[v];
  }
}

// ---------------- Kernel 6: energies = -logsumexp_k logits ---------------------
__global__ __launch_bounds__(256) void k_energy(const float* __restrict__ ws,
                                                float* __restrict__ out) {
  const int i = blockIdx.x * 256 + threadIdx.x;
  if (i >= GN) return;
  const float* lg = ws + WS_LOGITS + (size_t)i * GK;
  float m = lg[0];
#pragma unroll
  for (int k = 1; k < GK; k++) m = fmaxf(m, lg[k]);
  float s = 0.f;
#pragma unroll
  for (int k = 0; k < GK; k++) s += expf(lg[k] - m);
  out[i] = -(m + logf(s));
}

extern "C" void kernel_launch(void* const* d_in, const int* in_sizes, int n_in,
                              void* d_out, int out_size, void* d_ws, size_t ws_size,
                              hipStream_t stream) {
  (void)in_sizes; (void)n_in; (void)out_size; (void)ws_size;
  const float* z     = (const float*)d_in[0];   // [N, D] fp32
  const float* gamma = (const float*)d_in[1];   // [N, K] fp32
  float* ws  = (float*)d_ws;
  float* out = (float*)d_out;                   // [N] fp32

  k_stats_partial<<<64,   256, 0, stream>>>(z, gamma, ws);
  k_stats_reduce <<<1,    256, 0, stream>>>(ws);
  k_s2_wmma      <<<256,  256, 0, stream>>>(z, gamma, ws);
  k_chol         <<<16,   128, 0, stream>>>(ws);
  k_maha_wmma    <<<4096, 256, 0, stream>>>(z, ws);
  k_energy       <<<128,  256, 0, stream>>>(ws, out);
}